// BahdanauAttention_4166118277506
// MI455X (gfx1250) — compile-verified
//
#include <hip/hip_runtime.h>
#include <hip/hip_bf16.h>

typedef __attribute__((ext_vector_type(2))) float v2f;
typedef __attribute__((ext_vector_type(8))) float v8f;

#define B_ 4
#define Q_ 256
#define K_ 1024
#define H_ 128

// Unconditional: probe whether gfx1250 hardware tanh (v_tanh_f32, TRANS op)
// is exposed by this toolchain. If this fails to compile, fall back to an
// exp2-based tanh next round.
__device__ __forceinline__ float fast_tanh(float x) {
  return __builtin_amdgcn_tanhf(x);        // v_tanh_f32
}

__device__ __forceinline__ float fast_exp(float x) {
  return __builtin_amdgcn_exp2f(x * 1.4426950408889634f);  // v_exp_f32
}

// ---------------------------------------------------------------------------
// Y[m,h] = sum_d X[m,d] * W[h,d]   (torch Linear: x @ W^T), f32 WMMA 16x16x4
// grid = (M/16)*(H/16) blocks of 1 wave (32 threads).
// A[m,k] = X[m,k]; B[k,n] = W[n,k]  -> both are contiguous float2 loads.
// ---------------------------------------------------------------------------
__global__ __launch_bounds__(32) void proj_wmma(const float* __restrict__ X,
                                                const float* __restrict__ W,
                                                float* __restrict__ Y) {
  const int tilesN = H_ / 16;                 // 8
  const int tile = blockIdx.x;
  const int tm = tile / tilesN;
  const int tn = tile % tilesN;
  const int lane = threadIdx.x;               // 0..31 (wave32)
  const int half = lane >> 4;                 // lanes 16-31 hold K=2,3
  const int ln = lane & 15;

  const float* arow = X + (size_t)(tm * 16 + ln) * H_;
  const float* brow = W + (size_t)(tn * 16 + ln) * H_;

  v8f c = {};
#pragma unroll
  for (int kk = 0; kk < H_ / 4; ++kk) {       // 32 WMMAs, K = 128
    const int k0 = 4 * kk + 2 * half;
    v2f a = *(const v2f*)(arow + k0);         // A[m, k0..k0+1]
    v2f b = *(const v2f*)(brow + k0);         // W[n, k0..k0+1] == B[k,n]
    c = __builtin_amdgcn_wmma_f32_16x16x4_f32(false, a, false, b,
                                              (short)0, c, false, false);
  }

  // D layout: VGPR j -> M = j (lanes 0-15) / M = j+8 (lanes 16-31), N = ln
  float* yrow = Y + (size_t)(tm * 16 + 8 * half) * H_ + tn * 16 + ln;
#pragma unroll
  for (int j = 0; j < 8; ++j) yrow[(size_t)j * H_] = c[j];
}

// ---------------------------------------------------------------------------
// Fused additive score + mask + softmax.  One block per (b,q) row.
// score[k] = sum_h v[h]*tanh(qp[bq,h] + kp[b,k,h]) - 1e-9*(1-mask)
// ---------------------------------------------------------------------------
__global__ __launch_bounds__(256) void score_softmax(
    const float* __restrict__ qp, const float* __restrict__ kp,
    const int* __restrict__ mask, const float* __restrict__ v,
    float* __restrict__ attn) {
  __shared__ float qv[H_];
  __shared__ float vv[H_];
  __shared__ float sc[K_];
  __shared__ float red[256];

  const int bq = blockIdx.x;                  // 0..B*Q-1
  const int b = bq >> 8;                      // Q_ == 256
  const int tid = threadIdx.x;

  if (tid < H_) {
    qv[tid] = qp[(size_t)bq * H_ + tid];
    vv[tid] = v[tid];
  }
  __syncthreads();

  const float* kbase = kp + (size_t)b * K_ * H_;
  const int* mrow = mask + (size_t)bq * K_;

  float lmax = -3.402823466e38f;
#pragma unroll
  for (int kt = 0; kt < K_ / 256; ++kt) {
    const int k = kt * 256 + tid;
    const float4* krow = (const float4*)(kbase + (size_t)k * H_);
    float acc = 0.f;
#pragma unroll 8
    for (int h4 = 0; h4 < H_ / 4; ++h4) {
      const float4 kv = krow[h4];
      const int h = h4 * 4;
      acc += vv[h + 0] * fast_tanh(qv[h + 0] + kv.x);
      acc += vv[h + 1] * fast_tanh(qv[h + 1] + kv.y);
      acc += vv[h + 2] * fast_tanh(qv[h + 2] + kv.z);
      acc += vv[h + 3] * fast_tanh(qv[h + 3] + kv.w);
    }
    acc += -1e-9f * (1.0f - (float)mrow[k]);  // NEG_MASK_SCALE faithful to ref
    sc[k] = acc;
    lmax = fmaxf(lmax, acc);
  }

  red[tid] = lmax;
  __syncthreads();
  for (int s = 128; s > 0; s >>= 1) {
    if (tid < s) red[tid] = fmaxf(red[tid], red[tid + s]);
    __syncthreads();
  }
  const float mx = red[0];
  __syncthreads();

  float lsum = 0.f;
#pragma unroll
  for (int kt = 0; kt < K_ / 256; ++kt) {
    const int k = kt * 256 + tid;
    const float e = fast_exp(sc[k] - mx);
    sc[k] = e;
    lsum += e;
  }
  red[tid] = lsum;
  __syncthreads();
  for (int s = 128; s > 0; s >>= 1) {
    if (tid < s) red[tid] += red[tid + s];
    __syncthreads();
  }
  const float inv = 1.0f / red[0];

  float* arow = attn + (size_t)bq * K_;
#pragma unroll
  for (int kt = 0; kt < K_ / 256; ++kt) {
    const int k = kt * 256 + tid;
    arow[k] = sc[k] * inv;
  }
}

// ---------------------------------------------------------------------------
// ctx[b,q,h] = sum_k attn[b,q,k] * V[b,k,h]   — f32 WMMA, K = 1024
// grid = B * (Q/16) * (H/16) = 512 single-wave blocks, 256 WMMAs each.
// ---------------------------------------------------------------------------
__global__ __launch_bounds__(32) void ctx_wmma(const float* __restrict__ attn,
                                               const float* __restrict__ V,
                                               float* __restrict__ ctx) {
  const int tilesH = H_ / 16;                 // 8
  const int tilesQ = Q_ / 16;                 // 16
  const int id = blockIdx.x;
  const int th = id % tilesH;
  const int tq = (id / tilesH) % tilesQ;
  const int b = id / (tilesH * tilesQ);
  const int lane = threadIdx.x;
  const int half = lane >> 4;
  const int ln = lane & 15;

  const float* arow = attn + ((size_t)b * Q_ + tq * 16 + ln) * K_;
  const float* vbase = V + (size_t)b * K_ * H_ + th * 16 + ln;

  v8f c = {};
#pragma unroll 8
  for (int kk = 0; kk < K_ / 4; ++kk) {
    const int k0 = 4 * kk + 2 * half;
    v2f a = *(const v2f*)(arow + k0);         // attn[q, k0..k0+1] contiguous
    v2f bb;
    bb.x = vbase[(size_t)k0 * H_];            // V[k0,   h]
    bb.y = vbase[(size_t)(k0 + 1) * H_];      // V[k0+1, h]
    c = __builtin_amdgcn_wmma_f32_16x16x4_f32(false, a, false, bb,
                                              (short)0, c, false, false);
  }

  float* crow = ctx + ((size_t)b * Q_ + tq * 16 + 8 * half) * H_ + th * 16 + ln;
#pragma unroll
  for (int j = 0; j < 8; ++j) crow[(size_t)j * H_] = c[j];
}

// ---------------------------------------------------------------------------
extern "C" void kernel_launch(void* const* d_in, const int* in_sizes, int n_in,
                              void* d_out, int out_size, void* d_ws, size_t ws_size,
                              hipStream_t stream) {
  const float* query = (const float*)d_in[0];   // (B,Q,H)
  const float* key   = (const float*)d_in[1];   // (B,K,H)
  const float* value = (const float*)d_in[2];   // (B,K,H)
  const int*   mask  = (const int*)  d_in[3];   // (B,Q,K)
  const float* Wq    = (const float*)d_in[4];   // (H,H)
  const float* Wk    = (const float*)d_in[5];   // (H,H)
  const float* v     = (const float*)d_in[6];   // (H)

  float* attn = (float*)d_out;                              // (B,Q,K)
  float* ctx  = (float*)d_out + (size_t)B_ * Q_ * K_;       // (B,Q,H)

  float* qp = (float*)d_ws;                                 // (B*Q, H) 512 KB
  float* kp = qp + (size_t)B_ * Q_ * H_;                    // (B*K, H) 2 MB

  // Projections (f32 WMMA 16x16x4)
  proj_wmma<<<(B_ * Q_ / 16) * (H_ / 16), 32, 0, stream>>>(query, Wq, qp); // 512
  proj_wmma<<<(B_ * K_ / 16) * (H_ / 16), 32, 0, stream>>>(key,   Wk, kp); // 2048

  // Fused additive score + mask + softmax (dominant: 134M v_tanh_f32)
  score_softmax<<<B_ * Q_, 256, 0, stream>>>(qp, kp, mask, v, attn);

  // Context GEMM (f32 WMMA, K = 1024)
  ctx_wmma<<<B_ * (Q_ / 16) * (H_ / 16), 32, 0, stream>>>(attn, value, ctx);
}